// BodyMeasurements_12086037971381
// MI455X (gfx1250) — compile-verified
//
#include <hip/hip_runtime.h>
#include <math.h>

#define DENSITY  985.0f
#define NUM_DIRS 720
#define BATCH    4

typedef __attribute__((ext_vector_type(2))) float v2f;
typedef __attribute__((ext_vector_type(8))) float v8f;

// ---------------------------------------------------------------------------
// Kernel 1: per-face signed tet volume, deterministic block partials.
// ---------------------------------------------------------------------------
__global__ void __launch_bounds__(256)
mass_kernel(const float* __restrict__ tris, float* __restrict__ partials,
            int F, int nblk) {
    __shared__ float red[256];
    const int b = blockIdx.y;
    const int f = blockIdx.x * blockDim.x + threadIdx.x;
    float vol = 0.0f;
    if (f < F) {
        const float* v = tris + ((size_t)b * F + f) * 9;
        float x0 = v[0], y0 = v[1], z0 = v[2];
        float x1 = v[3], y1 = v[4], z1 = v[5];
        float x2 = v[6], y2 = v[7], z2 = v[8];
        vol = -x2 * y1 * z0 + x1 * y2 * z0 + x2 * y0 * z1
              - x0 * y2 * z1 - x1 * y0 * z2 + x0 * y1 * z2;
    }
    red[threadIdx.x] = vol;
    __syncthreads();
    for (int s = 128; s > 0; s >>= 1) {
        if (threadIdx.x < s) red[threadIdx.x] += red[threadIdx.x + s];
        __syncthreads();
    }
    if (threadIdx.x == 0) partials[b * nblk + blockIdx.x] = red[0];
}

// ---------------------------------------------------------------------------
// Kernel 2: finalize mass, compute height + the 3 section heights h[m][b].
// ---------------------------------------------------------------------------
__device__ __forceinline__ float bary_y(const float* tris, int b, int F,
                                        int face, const float* bc) {
    const float* v = tris + ((size_t)b * F + face) * 9;
    return bc[0] * v[1] + bc[1] * v[4] + bc[2] * v[7];
}

__global__ void points_kernel(const float* __restrict__ tris,
                              const float* __restrict__ bc_head,
                              const float* __restrict__ bc_heel,
                              const float* __restrict__ bc_chest,
                              const float* __restrict__ bc_belly,
                              const float* __restrict__ bc_hips,
                              const int* __restrict__ i_head,
                              const int* __restrict__ i_heel,
                              const int* __restrict__ i_chest,
                              const int* __restrict__ i_belly,
                              const int* __restrict__ i_hips,
                              const float* __restrict__ partials,
                              float* __restrict__ wsh,
                              float* __restrict__ out,
                              int F, int nblk) {
    const int b = threadIdx.x;
    if (b >= BATCH) return;
    float acc = 0.0f;
    for (int i = 0; i < nblk; ++i) acc += partials[b * nblk + i];
    out[0 * BATCH + b] = fabsf(acc) * (DENSITY / 6.0f);

    float yh = bary_y(tris, b, F, *i_head, bc_head);
    float yl = bary_y(tris, b, F, *i_heel, bc_heel);
    out[1 * BATCH + b] = fabsf(yh - yl);

    wsh[0 * BATCH + b] = bary_y(tris, b, F, *i_chest, bc_chest);
    wsh[1 * BATCH + b] = bary_y(tris, b, F, *i_belly, bc_belly);
    wsh[2 * BATCH + b] = bary_y(tris, b, F, *i_hips,  bc_hips);
}

// ---------------------------------------------------------------------------
// Kernel 3: WMMA-based projection + argmax per direction.
// grid = (NUM_DIRS/16, 12); block = 256 (8 waves). Each wave owns 16 dirs
// (B-matrix), strides over 16-point tiles (A-matrix), computes the 16x16
// projection tile with V_WMMA_F32_16X16X4_F32.
//
// Validity masking is folded INTO the WMMA via the otherwise-unused K=2 row:
//   B[K=2][*] = 1.0   (B VGPR0, lanes 16..31)
//   A[M][K=2] = valid(M) ? 0 : -1e30   (A VGPR0, lanes 16..31, M = lane-16)
// so D = x*cos + z*sin + bias. Running max starts at -1e25: every valid
// projection (|coords| ~ O(1)) beats it, every invalid candidate (~ -1e30)
// loses to it, and the all-invalid case leaves idx = 0 — exactly matching
// jnp.argmax over all -inf.
//
// Both lane-halves compute the SAME point (M = lane & 15): lanes 0..15 use
// the (x,z) locally for A's K=0/1 slots, lanes 16..31 use the locally
// computed bias for A's K=2 slot. The duplicate math costs no extra wave
// cycles (the halves would otherwise idle under EXEC) and the duplicate
// loads hit the same cachelines — but it removes the per-tile ds_bpermute +
// s_wait_dscnt from the WMMA's critical path.
//
// D layout: VGPR r, lane l -> M = r + 8*(l>=16) (point class), N = l%16 (dir).
// ---------------------------------------------------------------------------
#define SENTINEL (-1e25f)
#define BIGNEG   (-1e30f)

__global__ void __launch_bounds__(256)
argmax_kernel(const float* __restrict__ tris, const float* __restrict__ wsh,
              int* __restrict__ wsidx, int F) {
    const int mb      = blockIdx.y;        // m*4 + b
    const int b       = mb & 3;
    const int dirbase = blockIdx.x * 16;
    const int wave    = threadIdx.x >> 5;
    const int lane    = threadIdx.x & 31;
    const int lane16  = lane & 15;
    const float h     = wsh[mb];
    const bool  lo    = (lane < 16);

    // B matrix: lanes 0..15 -> (K=0,K=1) = (cos, sin); lanes 16..31 ->
    // (K=2,K=3) = (1, 0) so the bias column of A passes straight through.
    float dc = 1.0f, ds = 0.0f;
    if (lo) {
        int k = dirbase + lane;
        float theta = (float)k * (6.28318530717958647692f / (float)NUM_DIRS);
        if (k < NUM_DIRS) {
            dc = cosf(theta);
            ds = sinf(theta);
        }
    }
    v2f Bm = {dc, ds};

    float rmax[8];
    int   ridx[8];
#pragma unroll
    for (int r = 0; r < 8; ++r) { rmax[r] = SENTINEL; ridx[r] = 0; }

    const int P      = F * 3;
    const int ntiles = (P + 15) >> 4;
    const int mbase  = (lane >> 4) << 3;   // 0 for lanes 0..15, 8 for 16..31
    const float* tb  = tris + (size_t)b * F * 9;

    for (int t = wave; t < ntiles; t += 8) {
        // Every lane computes point M = lane & 15 of this tile (both halves
        // duplicate the work; see header comment).
        float px = 0.0f, pz = 0.0f;
        float bias = BIGNEG;               // invalid unless proven otherwise
        int   p = (t << 4) + lane16;
        if (p < P) {                       // wave-uniform except tail tile
            int f  = p / 3;
            int e  = p - f * 3;
            int e2 = (e == 2) ? 0 : e + 1;
            const float* vb = tb + (size_t)f * 9;
            float xi = vb[e * 3 + 0], yi = vb[e * 3 + 1], zi = vb[e * 3 + 2];
            float xj = vb[e2 * 3 + 0], yj = vb[e2 * 3 + 1], zj = vb[e2 * 3 + 2];
            bool valid = (yi - h) * (yj - h) < 0.0f;
            bias = valid ? 0.0f : BIGNEG;
            float denom = yj - yi;
            float safe  = (fabsf(denom) > 1e-12f) ? denom : 1e-12f;
            float tt    = fminf(fmaxf((h - yi) / safe, 0.0f), 1.0f);
            px = xi + tt * (xj - xi);
            pz = zi + tt * (zj - zi);
        }
        v2f Am;
        Am.x = lo ? px : bias;             // K=0 (x)  | K=2 (bias)
        Am.y = lo ? pz : 0.0f;             // K=1 (z)  | K=3 (0)
        v8f D = {0.f, 0.f, 0.f, 0.f, 0.f, 0.f, 0.f, 0.f};
        // EXEC is all ones here (no lane-divergent branch encloses this).
        D = __builtin_amdgcn_wmma_f32_16x16x4_f32(
                /*neg_a=*/false, Am, /*neg_b=*/false, Bm,
                /*c_mod=*/(short)0, D, /*reuse_a=*/false, /*reuse_b=*/false);

        const int base = (t << 4) + mbase;
#pragma unroll
        for (int r = 0; r < 8; ++r) {
            if (D[r] > rmax[r]) { rmax[r] = D[r]; ridx[r] = base + r; }
        }
    }

    // Reduce the 8 M-class registers within the lane.
    float bm = rmax[0];
    int   bi = ridx[0];
#pragma unroll
    for (int r = 1; r < 8; ++r)
        if (rmax[r] > bm) { bm = rmax[r]; bi = ridx[r]; }

    // Reduce lane halves (same direction N, M classes offset by 8). wave32.
    float om = __shfl_xor(bm, 16, 32);
    int   oi = __shfl_xor(bi, 16, 32);
    if (om > bm) { bm = om; bi = oi; }

    // Reduce across the 8 waves via LDS.
    __shared__ float smax[8][16];
    __shared__ int   sidx[8][16];
    if (lo) { smax[wave][lane] = bm; sidx[wave][lane] = bi; }
    __syncthreads();
    if (wave == 0 && lo) {
        for (int w = 1; w < 8; ++w)
            if (smax[w][lane] > bm) { bm = smax[w][lane]; bi = sidx[w][lane]; }
        int k = dirbase + lane;
        if (k < NUM_DIRS) wsidx[mb * NUM_DIRS + k] = bi;
    }
}

// ---------------------------------------------------------------------------
// Kernel 4: rebuild the 720 extreme points from indices, sum segment lengths.
// ---------------------------------------------------------------------------
__global__ void __launch_bounds__(256)
perim_kernel(const float* __restrict__ tris, const float* __restrict__ wsh,
             const int* __restrict__ wsidx, float* __restrict__ out, int F) {
    const int mb = blockIdx.x;  // 0..11
    const int m  = mb >> 2, b = mb & 3;
    __shared__ float ex[NUM_DIRS];
    __shared__ float ez[NUM_DIRS];
    __shared__ float red[256];
    const float h  = wsh[mb];
    const float* tb = tris + (size_t)b * F * 9;

    for (int k = threadIdx.x; k < NUM_DIRS; k += blockDim.x) {
        int p  = wsidx[mb * NUM_DIRS + k];
        int f  = p / 3;
        int e  = p - f * 3;
        int e2 = (e == 2) ? 0 : e + 1;
        const float* vb = tb + (size_t)f * 9;
        float xi = vb[e * 3 + 0], yi = vb[e * 3 + 1], zi = vb[e * 3 + 2];
        float xj = vb[e2 * 3 + 0], yj = vb[e2 * 3 + 1], zj = vb[e2 * 3 + 2];
        float denom = yj - yi;
        float safe  = (fabsf(denom) > 1e-12f) ? denom : 1e-12f;
        float tt    = fminf(fmaxf((h - yi) / safe, 0.0f), 1.0f);
        ex[k] = xi + tt * (xj - xi);
        ez[k] = zi + tt * (zj - zi);
    }
    __syncthreads();

    float s = 0.0f;
    for (int k = threadIdx.x; k < NUM_DIRS; k += blockDim.x) {
        int k2 = (k + 1 == NUM_DIRS) ? 0 : k + 1;
        float dx = ex[k] - ex[k2];
        float dz = ez[k] - ez[k2];
        s += sqrtf(dx * dx + dz * dz + 1e-20f);
    }
    red[threadIdx.x] = s;
    __syncthreads();
    for (int st = 128; st > 0; st >>= 1) {
        if (threadIdx.x < st) red[threadIdx.x] += red[threadIdx.x + st];
        __syncthreads();
    }
    if (threadIdx.x == 0) out[(2 + m) * BATCH + b] = red[0];
}

// ---------------------------------------------------------------------------
extern "C" void kernel_launch(void* const* d_in, const int* in_sizes, int n_in,
                              void* d_out, int out_size, void* d_ws, size_t ws_size,
                              hipStream_t stream) {
    const float* tris     = (const float*)d_in[0];
    const float* bc_head  = (const float*)d_in[1];
    const float* bc_heel  = (const float*)d_in[2];
    const float* bc_chest = (const float*)d_in[3];
    const float* bc_belly = (const float*)d_in[4];
    const float* bc_hips  = (const float*)d_in[5];
    const int*   i_head   = (const int*)d_in[6];
    const int*   i_heel   = (const int*)d_in[7];
    const int*   i_chest  = (const int*)d_in[8];
    const int*   i_belly  = (const int*)d_in[9];
    const int*   i_hips   = (const int*)d_in[10];
    float* out = (float*)d_out;

    const int F    = in_sizes[0] / (BATCH * 9);
    const int nblk = (F + 255) / 256;

    // Workspace layout:
    //   [0 .. 11]                 floats : h[m][b]      (section heights)
    //   [16 .. 16+4*nblk)         floats : mass partials
    //   byte offset 4096 onward   ints   : argmax indices [12][NUM_DIRS]
    float* wsh      = (float*)d_ws;
    float* partials = (float*)d_ws + 16;
    int*   wsidx    = (int*)((char*)d_ws + 4096);

    mass_kernel<<<dim3(nblk, BATCH), 256, 0, stream>>>(tris, partials, F, nblk);

    points_kernel<<<1, 32, 0, stream>>>(tris, bc_head, bc_heel, bc_chest,
                                        bc_belly, bc_hips, i_head, i_heel,
                                        i_chest, i_belly, i_hips,
                                        partials, wsh, out, F, nblk);

    argmax_kernel<<<dim3((NUM_DIRS + 15) / 16, 12), 256, 0, stream>>>(
        tris, wsh, wsidx, F);

    perim_kernel<<<12, 256, 0, stream>>>(tris, wsh, wsidx, out, F);
}